// Temp_conv_18425409699926
// MI455X (gfx1250) — compile-verified
//
#include <hip/hip_runtime.h>
#include <hip/hip_bf16.h>

// ---------------------------------------------------------------------------
// Shapes: B=8, T=16, N=32, C=512, n = T*N = 512, KSZ=7, conv out len = 10
// Pipeline:
//   K1 normalize : normB[b,n,c] = feat/||feat||,  normW = normB * diag(W)[c]
//   K2 WMMA GEMM : A[b] = normW[b] (512x512) @ normB[b]^T   (fp32 16x16x4)
//                  32x32 tile per wave (2x2 register blocking)
//   K3 argmax    : top_A[b,n,t] = argmax_{nd<32} A[b,n,t*32+nd] (first-max tie)
//   K4 fused     : gather + depthwise conv7 + LayerNorm(C) + LeakyReLU + mean
// ---------------------------------------------------------------------------

typedef __attribute__((ext_vector_type(2))) float v2f;
typedef __attribute__((ext_vector_type(8))) float v8f;

#define CC 512
#define NROWS 4096          // B * n = 8 * 512

// --- K1: row L2 normalize, also pre-scale by diag(W) ------------------------
__global__ __launch_bounds__(256) void k_normalize(
    const float* __restrict__ feat, const float* __restrict__ W,
    float* __restrict__ normW, float* __restrict__ normB)
{
    const int row = blockIdx.x;           // b*512 + n
    const int tid = threadIdx.x;
    const float* f = feat + (size_t)row * CC;
    float x0 = f[tid], x1 = f[tid + 256];

    __shared__ float red[256];
    red[tid] = x0 * x0 + x1 * x1;
    __syncthreads();
    for (int off = 128; off > 0; off >>= 1) {
        if (tid < off) red[tid] += red[tid + off];
        __syncthreads();
    }
    const float rn = 1.0f / fmaxf(sqrtf(red[0]), 1e-12f);

    const float wd0 = W[(size_t)tid * (CC + 1)];          // W[tid][tid]
    const float wd1 = W[(size_t)(tid + 256) * (CC + 1)];  // W[tid+256][tid+256]

    const size_t base = (size_t)row * CC;
    normB[base + tid]       = x0 * rn;
    normB[base + tid + 256] = x1 * rn;
    normW[base + tid]       = x0 * rn * wd0;
    normW[base + tid + 256] = x1 * rn * wd1;
}

// --- K2: per-batch GEMM  A = normW @ normB^T  via V_WMMA_F32_16X16X4_F32 ----
// Both operands row-major 512x512, contraction over the column index.
// Each wave owns a 32x32 output tile = 2x2 grid of 16x16 WMMA tiles, so each
// A/B fragment load feeds two WMMAs (1 b64 load per wmma instead of 2).
// grid = (32, 8): 8 waves/block, 256 tiles of 32x32 per batch.
__global__ __launch_bounds__(256) void k_gemm_wmma(
    const float* __restrict__ Am, const float* __restrict__ Bm,
    float* __restrict__ D)
{
    const int lane = threadIdx.x & 31;
    const int wave = threadIdx.x >> 5;
    const int tile = blockIdx.x * 8 + wave;        // 0..255
    const int b    = blockIdx.y;
    const int m0   = (tile >> 4) * 32;
    const int n0   = (tile & 15) * 32;
    const int l    = lane & 15;
    const int h    = lane >> 4;

    const float* Ab = Am + (size_t)b * CC * CC;
    const float* Bb = Bm + (size_t)b * CC * CC;
    float*       Db = D  + (size_t)b * CC * CC;

    // Fragment layouts (CDNA5 ISA):
    //   A frag (16x4): a[v] = A[m0+l][k0 + v + 2h]  -> contiguous float2
    //   B frag (4x16): b[v] = B[k0+v+2h][n0+l] = normB[n0+l][k0+v+2h] -> float2
    const float* arow0 = Ab + (size_t)(m0 + l) * CC + 2 * h;
    const float* arow1 = arow0 + (size_t)16 * CC;
    const float* brow0 = Bb + (size_t)(n0 + l) * CC + 2 * h;
    const float* brow1 = brow0 + (size_t)16 * CC;

    v8f acc00 = {}, acc01 = {}, acc10 = {}, acc11 = {};
#pragma unroll 2
    for (int k0 = 0; k0 < CC; k0 += 4) {
        v2f a0 = *(const v2f*)(arow0 + k0);
        v2f a1 = *(const v2f*)(arow1 + k0);
        v2f b0 = *(const v2f*)(brow0 + k0);
        v2f b1 = *(const v2f*)(brow1 + k0);
        acc00 = __builtin_amdgcn_wmma_f32_16x16x4_f32(false, a0, false, b0,
                                                      (short)0, acc00, false, false);
        acc01 = __builtin_amdgcn_wmma_f32_16x16x4_f32(false, a0, false, b1,
                                                      (short)0, acc01, false, false);
        acc10 = __builtin_amdgcn_wmma_f32_16x16x4_f32(false, a1, false, b0,
                                                      (short)0, acc10, false, false);
        acc11 = __builtin_amdgcn_wmma_f32_16x16x4_f32(false, a1, false, b1,
                                                      (short)0, acc11, false, false);
    }

    // D frag: VGPR v holds row (mbase + v + 8h), col (nbase + l)
    float* d00 = Db + (size_t)(m0 + 8 * h) * CC + n0 + l;
#pragma unroll
    for (int v = 0; v < 8; ++v) {
        d00[(size_t)v * CC]                      = acc00[v];   // rows m0..,  cols n0..
        d00[(size_t)v * CC + 16]                 = acc01[v];   // rows m0..,  cols n0+16..
        d00[(size_t)(v + 16) * CC]               = acc10[v];   // rows m0+16, cols n0..
        d00[(size_t)(v + 16) * CC + 16]          = acc11[v];   // rows m0+16, cols n0+16
    }
}

// --- K3: argmax over nd (32 consecutive columns) -> first-max index ---------
// One wave per group; 65536 groups total; 8 waves per block -> 8192 blocks.
__global__ __launch_bounds__(256) void k_argmax(
    const float* __restrict__ A, int* __restrict__ topA)
{
    const int lane = threadIdx.x & 31;
    const int wave = threadIdx.x >> 5;
    const int g = blockIdx.x * 8 + wave;          // ((b*512 + n)*16 + t)

    float v = A[(size_t)g * 32 + lane];
    int   idx = lane;
#pragma unroll
    for (int off = 16; off > 0; off >>= 1) {
        float ov = __shfl_xor(v,   off, 32);
        int   oi = __shfl_xor(idx, off, 32);
        if (ov > v || (ov == v && oi < idx)) { v = ov; idx = oi; }
    }
    if (lane == 0) topA[g] = idx;
}

// --- K4: gather + depthwise conv7 + LayerNorm + LeakyReLU + mean ------------
// One block per (b, n) row; thread tid handles channels tid and tid+256.
__global__ __launch_bounds__(256) void k_conv_ln_act_mean(
    const float* __restrict__ lf, const int* __restrict__ topA,
    const float* __restrict__ cw, const float* __restrict__ cb,
    const float* __restrict__ gamma, const float* __restrict__ beta,
    float* __restrict__ out)
{
    const int row = blockIdx.x;          // b*512 + n
    const int b   = row >> 9;
    const int tid = threadIdx.x;
    const int c0 = tid, c1 = tid + 256;

    __shared__ int    sidx[16];
    __shared__ float2 red[256];
    if (tid < 16) sidx[tid] = topA[row * 16 + tid];
    __syncthreads();

    // Gather top_feat[b,n,t,c] = local_feat[b, t, top_A[b,n,t], c]
    float f0[16], f1[16];
#pragma unroll
    for (int t = 0; t < 16; ++t) {
        const float* p = lf + (size_t)((b * 16 + t) * 32 + sidx[t]) * CC;
        f0[t] = p[c0];
        f1[t] = p[c1];
    }

    float w0[7], w1[7];
#pragma unroll
    for (int k = 0; k < 7; ++k) { w0[k] = cw[c0 * 7 + k]; w1[k] = cw[c1 * 7 + k]; }
    const float bias0 = cb[c0], bias1 = cb[c1];
    const float g0 = gamma[c0], g1 = gamma[c1];
    const float be0 = beta[c0], be1 = beta[c1];

    // Depthwise conv along t, VALID -> 10 outputs per channel
    float y0[10], y1[10];
#pragma unroll
    for (int j = 0; j < 10; ++j) {
        float s0 = bias0, s1 = bias1;
#pragma unroll
        for (int k = 0; k < 7; ++k) {
            s0 = fmaf(f0[j + k], w0[k], s0);
            s1 = fmaf(f1[j + k], w1[k], s1);
        }
        y0[j] = s0; y1[j] = s1;
    }

    // Per-j LayerNorm over C=512, LeakyReLU, then mean over the 10 positions
    float acc0 = 0.f, acc1 = 0.f;
    for (int j = 0; j < 10; ++j) {
        red[tid] = make_float2(y0[j] + y1[j], y0[j] * y0[j] + y1[j] * y1[j]);
        __syncthreads();
        for (int off = 128; off > 0; off >>= 1) {
            if (tid < off) {
                red[tid].x += red[tid + off].x;
                red[tid].y += red[tid + off].y;
            }
            __syncthreads();
        }
        const float mu  = red[0].x * (1.0f / 512.0f);
        const float var = red[0].y * (1.0f / 512.0f) - mu * mu;
        __syncthreads();
        const float rstd = rsqrtf(var + 1e-5f);
        const float l0 = (y0[j] - mu) * rstd * g0 + be0;
        const float l1 = (y1[j] - mu) * rstd * g1 + be1;
        acc0 += (l0 >= 0.f) ? l0 : 0.01f * l0;
        acc1 += (l1 >= 0.f) ? l1 : 0.01f * l1;
    }

    out[(size_t)row * CC + c0] = acc0 * 0.1f;
    out[(size_t)row * CC + c1] = acc1 * 0.1f;
}

// ---------------------------------------------------------------------------
extern "C" void kernel_launch(void* const* d_in, const int* in_sizes, int n_in,
                              void* d_out, int out_size, void* d_ws, size_t ws_size,
                              hipStream_t stream) {
    (void)in_sizes; (void)n_in; (void)out_size; (void)ws_size;

    const float* lf    = (const float*)d_in[0];  // local_feat (8,16,32,512)
    // d_in[1] global_feat, d_in[2] pos: unused by the reference
    const float* W     = (const float*)d_in[3];  // (512,512) -> only diagonal used
    const float* cw    = (const float*)d_in[4];  // conv_w (512,1,7)
    const float* cb    = (const float*)d_in[5];  // conv_b (512)
    const float* gamma = (const float*)d_in[6];  // ln_gamma (512)
    const float* beta  = (const float*)d_in[7];  // ln_beta (512)
    float* out = (float*)d_out;

    char* ws = (char*)d_ws;
    const size_t MAT = (size_t)8 * CC * CC * sizeof(float);  // 8 MiB per stage
    float* normW = (float*)(ws);
    float* normB = (float*)(ws + MAT);
    float* Abuf  = (float*)(ws + 2 * MAT);
    int*   topA  = (int*)  (ws + 3 * MAT);

    k_normalize<<<NROWS, 256, 0, stream>>>(lf, W, normW, normB);
    k_gemm_wmma<<<dim3(32, 8), 256, 0, stream>>>(normW, normB, Abuf);
    k_argmax<<<8192, 256, 0, stream>>>(Abuf, topA);
    k_conv_ln_act_mean<<<NROWS, 256, 0, stream>>>(lf, topA, cw, cb, gamma, beta, out);
}